// GINLocator_32341103739012
// MI455X (gfx1250) — compile-verified
//
#include <hip/hip_runtime.h>

typedef __attribute__((ext_vector_type(2))) float v2f;
typedef __attribute__((ext_vector_type(8))) float v8f;

#define DD 64
#define LSTR 68  // padded LDS row stride (floats): 8B-aligned float2 slots, conflict-free banks

__device__ __forceinline__ float swish_f(float x) {
  return x / (1.0f + __expf(-x));
}

// ---------------------------------------------------------------- embedding
__global__ __launch_bounds__(256) void gin_embed(const int* __restrict__ state,
                                                 const float* __restrict__ emb,
                                                 float* __restrict__ h0, int n_nodes) {
  int t = blockIdx.x * 256 + threadIdx.x;
  if (t >= n_nodes * DD) return;
  int node = t >> 6, d = t & 63;
  h0[t] = emb[state[node] * DD + d];
}

__global__ __launch_bounds__(256) void gin_zero(float* __restrict__ p, int n) {
  int t = blockIdx.x * 256 + threadIdx.x;
  if (t < n) p[t] = 0.0f;
}

// ------------------------------------------- x = swish(Hin @ W + b), via f32 WMMA
// One wave per 16-node strip; strip's 16x64 A tile and the 64x64 W (transposed)
// staged in LDS. K=64 swept with 16 chained V_WMMA_F32_16X16X4_F32 per N-tile.
__global__ __launch_bounds__(256) void gin_fc_swish_wmma(
    const float* __restrict__ Hin, int hin_stride,
    const float* __restrict__ W,      // [64,64] row-major (k,n)
    const float* __restrict__ bias,   // [64]
    float* __restrict__ X,            // [N,64]  (scatter source)
    float* __restrict__ Hout, int hout_stride,  // layer slot inside H
    int n_nodes) {
  __shared__ __align__(16) float sW[DD * LSTR];       // W^T: sW[n*LSTR + k]
  __shared__ __align__(16) float sA[8][16 * LSTR];    // per-wave A strips

  const int tid  = threadIdx.x;
  const int wave = tid >> 5;
  const int lane = tid & 31;
  const int l16  = lane & 15;
  const int half = lane >> 4;

  // cooperative load+transpose of the 64x64 weight block
  for (int i = tid; i < DD * DD; i += 256) {
    int k = i >> 6, n = i & 63;
    sW[n * LSTR + k] = W[i];
  }

  const int strip = blockIdx.x * 8 + wave;
  const int row0  = strip * 16;
  const bool valid = (row0 < n_nodes);
  if (valid) {
    float* myA = sA[wave];
    for (int i = lane; i < 16 * DD; i += 32) {
      int r = i >> 6, k = i & 63;
      int row = row0 + r;
      myA[r * LSTR + k] = (row < n_nodes) ? Hin[(size_t)row * hin_stride + k] : 0.0f;
    }
  }
  __syncthreads();
  if (!valid) return;

  const float* myA = sA[wave];
#pragma unroll
  for (int nt = 0; nt < 4; ++nt) {
    const int col = nt * 16 + l16;
    v8f acc = {};
#pragma unroll
    for (int k0 = 0; k0 < 16; ++k0) {
      const int kk = 4 * k0 + 2 * half;
      // A 16x4 f32 layout: lanes 0-15 hold K=4*k0,4*k0+1 ; lanes 16-31 hold K=4*k0+2,4*k0+3
      v2f a = *(const v2f*)(myA + l16 * LSTR + kk);
      v2f b = *(const v2f*)(sW + col * LSTR + kk);
      acc = __builtin_amdgcn_wmma_f32_16x16x4_f32(
          /*neg_a=*/false, a, /*neg_b=*/false, b,
          /*c_mod=*/(short)0, acc, /*reuse_a=*/false, /*reuse_b=*/false);
    }
    const float bc = bias[col];
#pragma unroll
    for (int r = 0; r < 8; ++r) {
      int row = row0 + r + 8 * half;  // C/D layout: lanes 0-15 -> M=r, lanes 16-31 -> M=r+8
      if (row < n_nodes) {
        float s = swish_f(acc[r] + bc);
        X[(size_t)row * DD + col]            = s;
        Hout[(size_t)row * hout_stride + col] = s;
      }
    }
  }
}

// ---------------------------------------- Hout[dst] += X[src]  (GIN aggregation)
// 16 threads per edge: coalesced 256B float4 gather + 4 hardware f32 atomics.
__global__ __launch_bounds__(256) void gin_edge_scatter(
    const float* __restrict__ X, const int* __restrict__ src,
    const int* __restrict__ dst, float* __restrict__ Hout, int hstride, int n_edges) {
  int t = blockIdx.x * 256 + threadIdx.x;
  int e = t >> 4;
  if (e >= n_edges) return;
  int c = (t & 15) * 4;
  int s = src[e], d = dst[e];
  const float4 v = *(const float4*)(X + (size_t)s * DD + c);
  float* p = Hout + (size_t)d * hstride + c;
  unsafeAtomicAdd(p + 0, v.x);
  unsafeAtomicAdd(p + 1, v.y);
  unsafeAtomicAdd(p + 2, v.z);
  unsafeAtomicAdd(p + 3, v.w);
}

// ------------------ fused: z[g] += H[n]  and  logits[n] = swish(H[n]·W_logit + b)
// One wave per node; single pass over the 256-wide row.
__global__ __launch_bounds__(256) void gin_pool_logit(
    const float* __restrict__ H, const int* __restrict__ n2g,
    const float* __restrict__ Wl, const float* __restrict__ bl,
    float* __restrict__ z, float* __restrict__ logits, int n_nodes, int LD) {
  int gt = blockIdx.x * 256 + threadIdx.x;
  int node = gt >> 5;
  int lane = threadIdx.x & 31;
  if (node >= n_nodes) return;
  int g = n2g[node];
  const float* hrow = H + (size_t)node * LD;
  float* zrow = z + (size_t)g * LD;
  float dot = 0.0f;
  for (int c = lane * 4; c < LD; c += 128) {
    float4 hv = *(const float4*)(hrow + c);
    float4 wv = *(const float4*)(Wl + c);
    dot += hv.x * wv.x + hv.y * wv.y + hv.z * wv.z + hv.w * wv.w;
    unsafeAtomicAdd(zrow + c + 0, hv.x);
    unsafeAtomicAdd(zrow + c + 1, hv.y);
    unsafeAtomicAdd(zrow + c + 2, hv.z);
    unsafeAtomicAdd(zrow + c + 3, hv.w);
  }
#pragma unroll
  for (int off = 16; off > 0; off >>= 1) dot += __shfl_xor(dot, off, 32);
  if (lane == 0) logits[node] = swish_f(dot + bl[0]);
}

// ------------------------------------------ values[g] = swish(z[g]·W_val + b)
__global__ __launch_bounds__(256) void gin_values(
    const float* __restrict__ z, const float* __restrict__ Wv,
    const float* __restrict__ bv, float* __restrict__ values, int n_graphs, int LD) {
  int gt = blockIdx.x * 256 + threadIdx.x;
  int g = gt >> 5;
  int lane = threadIdx.x & 31;
  if (g >= n_graphs) return;
  const float* zrow = z + (size_t)g * LD;
  float dot = 0.0f;
  for (int c = lane * 4; c < LD; c += 128) {
    float4 zv = *(const float4*)(zrow + c);
    float4 wv = *(const float4*)(Wv + c);
    dot += zv.x * wv.x + zv.y * wv.y + zv.z * wv.z + zv.w * wv.w;
  }
#pragma unroll
  for (int off = 16; off > 0; off >>= 1) dot += __shfl_xor(dot, off, 32);
  if (lane == 0) values[g] = swish_f(dot + bv[0]);
}

extern "C" void kernel_launch(void* const* d_in, const int* in_sizes, int n_in,
                              void* d_out, int out_size, void* d_ws, size_t ws_size,
                              hipStream_t stream) {
  const int*   state   = (const int*)d_in[0];
  const int*   src     = (const int*)d_in[1];
  const int*   dst     = (const int*)d_in[2];
  const int*   n2g     = (const int*)d_in[3];
  // d_in[4] = n_graphs scalar (device); G derived from out_size instead
  const float* emb     = (const float*)d_in[5];
  const float* W_fc    = (const float*)d_in[6];
  const float* b_fc    = (const float*)d_in[7];
  const float* W_logit = (const float*)d_in[8];
  const float* b_logit = (const float*)d_in[9];
  const float* W_val   = (const float*)d_in[10];
  const float* b_val   = (const float*)d_in[11];

  const int N = in_sizes[0];
  const int E = in_sizes[1];
  const int G = out_size - N;
  const int L = in_sizes[6] / (DD * DD);
  const int LD = L * DD;

  float* h0     = (float*)d_ws;              // [N,64]
  float* X      = h0 + (size_t)N * DD;       // [N,64]
  float* Hbuf   = X + (size_t)N * DD;        // [N,L*64]
  float* z      = Hbuf + (size_t)N * LD;     // [G,L*64]
  float* logits = (float*)d_out;             // [N]
  float* values = logits + N;                // [G]

  gin_embed<<<(N * DD + 255) / 256, 256, 0, stream>>>(state, emb, h0, N);
  gin_zero<<<(G * LD + 255) / 256, 256, 0, stream>>>(z, G * LD);

  const int strips = (N + 15) / 16;
  const int fc_blocks = (strips + 7) / 8;
  const long scat_threads = (long)E * 16;
  const int scat_blocks = (int)((scat_threads + 255) / 256);

  for (int l = 0; l < L; ++l) {
    const float* Hin = (l == 0) ? h0 : (Hbuf + (size_t)(l - 1) * DD);
    const int hin_stride = (l == 0) ? DD : LD;
    gin_fc_swish_wmma<<<fc_blocks, 256, 0, stream>>>(
        Hin, hin_stride, W_fc + (size_t)l * DD * DD, b_fc + (size_t)l * DD,
        X, Hbuf + (size_t)l * DD, LD, N);
    gin_edge_scatter<<<scat_blocks, 256, 0, stream>>>(
        X, src, dst, Hbuf + (size_t)l * DD, LD, E);
  }

  gin_pool_logit<<<(int)(((long)N * 32 + 255) / 256), 256, 0, stream>>>(
      Hbuf, n2g, W_logit, b_logit, z, logits, N, LD);
  gin_values<<<(G * 32 + 255) / 256, 256, 0, stream>>>(z, W_val, b_val, values, G, LD);
}